// PFDetLoss_4380866642088
// MI455X (gfx1250) — compile-verified
//
#include <hip/hip_runtime.h>
#include <math.h>

#define BLOCK 256
#define MAXHW 2704   // 52*52, largest scale
#define NGT   32
#define FEPS  1e-7f

typedef float v2f __attribute__((ext_vector_type(2)));
typedef float v8f __attribute__((ext_vector_type(8)));

// Wave32 full reduction via the matrix pipe:
//   D = A(16x4) * ones(4x16) + 0, with A[m,0]=x[lane<16], A[m,2]=x[lane>=16]
//   => D[m][n] = x[m] + x[m+16] (independent of n).
// Summing the 8 D VGPRs per lane yields:
//   lanes 0..15 : sum over {0..7, 16..23};  lanes 16..31 : sum over {8..15, 24..31}
// so lanes 0 and 16 together hold the full wave sum (exact fp32).
__device__ __forceinline__ float wave_partial_sum(float x) {
#if __has_builtin(__builtin_amdgcn_wmma_f32_16x16x4_f32)
  v2f a; a[0] = x;    a[1] = 0.0f;
  v2f b; b[0] = 1.0f; b[1] = 1.0f;
  v8f c = {};
  v8f d = __builtin_amdgcn_wmma_f32_16x16x4_f32(
      /*neg_a=*/false, a, /*neg_b=*/false, b,
      /*c_mod=*/(short)0, c, /*reuse_a=*/false, /*reuse_b=*/false);
  return d[0] + d[1] + d[2] + d[3] + d[4] + d[5] + d[6] + d[7];
#else
  return x;  // fallback: caller's (tid&15)==0 guard is widened below
#endif
}

#if __has_builtin(__builtin_amdgcn_wmma_f32_16x16x4_f32)
#define RED_GUARD(tid) (((tid) & 15) == 0)   // lanes 0 and 16 of each wave
#else
#define RED_GUARD(tid) (1)                    // every thread adds its own partial
#endif

__global__ void det_loss_init(float* __restrict__ acc) {
  if (threadIdx.x < 4) acc[threadIdx.x] = 0.0f;
}

__global__ __launch_bounds__(BLOCK) void det_loss_main(
    const float* __restrict__ pred0, const float* __restrict__ pred1,
    const float* __restrict__ pred2, const float* __restrict__ gt,
    float* __restrict__ acc) {
  __shared__ int   s_idx[MAXHW];
  __shared__ float s_gt[NGT * 4];
  __shared__ float s_acc[3];

  const int b     = blockIdx.x;
  const int scale = blockIdx.y;
  const float* pred;
  int W; float strd;
  if (scale == 0)      { pred = pred0; W = 52; strd = 8.0f;  }
  else if (scale == 1) { pred = pred1; W = 26; strd = 16.0f; }
  else                 { pred = pred2; W = 13; strd = 32.0f; }
  const int   HW = W * W;
  const float Sf = 416.0f / strd;   // grid size as float (== W)
  const float sc = strd / 416.0f;   // cell size in normalized coords
  const int   tid = threadIdx.x;
  const float* __restrict__ base = pred + (size_t)b * 5 * HW;

  // ---- phase 0: LDS init ----
  for (int j = tid; j < HW; j += BLOCK) s_idx[j] = -1;
  if (tid < NGT * 4) s_gt[tid] = gt[(size_t)b * NGT * 4 + tid];
  if (tid < 3) s_acc[tid] = 0.0f;
  __syncthreads();

  // ---- phase 1: scatter-max of gt index into grid (ds_max_i32) ----
  if (tid < 3 * NGT) {
    const int n   = tid & (NGT - 1);
    const int var = tid >> 5;  // 0: center, 1: x-neighbor, 2: y-neighbor
    const float gx = s_gt[n * 4 + 0] * Sf;
    const float gy = s_gt[n * 4 + 1] * Sf;
    int col = (int)gx; col = col < 0 ? 0 : (col > W - 1 ? W - 1 : col);
    int row = (int)gy; row = row < 0 ? 0 : (row > W - 1 ? W - 1 : row);
    const float ox = gx - (float)col;
    const float oy = gy - (float)row;
    int r = row, c = col; bool valid = true;
    if (var == 1) {
      if (ox < 0.5f) { c = col - 1; valid = (col > 0); }
      else           { c = col + 1; valid = (col < W - 1); }
    } else if (var == 2) {
      if (oy < 0.5f) { r = row - 1; valid = (row > 0); }
      else           { r = row + 1; valid = (row < W - 1); }
    }
    if (valid) atomicMax(&s_idx[r * W + c], n);
  }
  __syncthreads();

  // ---- phase 2: per-cell loss; box planes touched only for positive cells ----
  float bce_t = 0.0f, box_t = 0.0f, npos_t = 0.0f;
  const float c4pi2 = 0.40528473456935108577f;  // 4 / pi^2
  for (int j = tid; j < HW; j += BLOCK) {
    const float l0 = base[j];
    float obj = 0.0f;
    const int g = s_idx[j];
    if (g >= 0) {
      const float l1 = base[HW + j];
      const float l2 = base[2 * HW + j];
      const float l3 = base[3 * HW + j];
      const float l4 = base[4 * HW + j];
      const float ccol = (float)(j % W);
      const float crow = (float)(j / W);
      const float sg1 = 1.0f / (1.0f + expf(-l1));
      const float sg2 = 1.0f / (1.0f + expf(-l2));
      const float pcx = (sg1 * 2.0f - 0.5f + ccol) * sc;
      const float pcy = (sg2 * 2.0f - 0.5f + crow) * sc;
      const float pw = expf(fminf(fmaxf(l3, -5.0f), 5.0f)) * sc;
      const float ph = expf(fminf(fmaxf(l4, -5.0f), 5.0f)) * sc;
      const float tcx = s_gt[g * 4 + 0], tcy = s_gt[g * 4 + 1];
      const float tw  = s_gt[g * 4 + 2], th  = s_gt[g * 4 + 3];
      const float b1x1 = pcx - pw * 0.5f, b1y1 = pcy - ph * 0.5f;
      const float b1x2 = pcx + pw * 0.5f, b1y2 = pcy + ph * 0.5f;
      const float b2x1 = tcx - tw * 0.5f, b2y1 = tcy - th * 0.5f;
      const float b2x2 = tcx + tw * 0.5f, b2y2 = tcy + th * 0.5f;
      const float w1 = b1x2 - b1x1, h1 = b1y2 - b1y1;
      const float w2 = b2x2 - b2x1, h2 = b2y2 - b2y1;
      float iw = fminf(b1x2, b2x2) - fmaxf(b1x1, b2x1); iw = iw > 0.0f ? iw : 0.0f;
      float ih = fminf(b1y2, b2y2) - fmaxf(b1y1, b2y1); ih = ih > 0.0f ? ih : 0.0f;
      const float inter = iw * ih;
      const float uni   = w1 * h1 + w2 * h2 - inter + FEPS;
      const float iou   = inter / uni;
      const float cw = fmaxf(b1x2, b2x2) - fminf(b1x1, b2x1);
      const float ch = fmaxf(b1y2, b2y2) - fminf(b1y1, b2y1);
      const float c2 = cw * cw + ch * ch + FEPS;
      const float dx = b2x1 + b2x2 - b1x1 - b1x2;
      const float dy = b2y1 + b2y2 - b1y1 - b1y2;
      const float rho2 = (dx * dx + dy * dy) * 0.25f;
      const float dat = atanf(w2 / (h2 + FEPS)) - atanf(w1 / (h1 + FEPS));
      const float v = c4pi2 * dat * dat;
      const float alpha = v / (v - iou + 1.0f + FEPS);
      const float ciou = 1.0f - iou + rho2 / c2 + alpha * v;
      box_t  += fmaxf(ciou, 0.0f);
      npos_t += 1.0f;
      const float ap = fmaxf(pw * ph, FEPS);
      const float at = fmaxf(tw * th, FEPS);
      const float iou2 = inter / (ap + at - inter + FEPS);
      obj = fminf(fmaxf(iou2, 0.0f), 1.0f);
    }
    bce_t += fmaxf(l0, 0.0f) - l0 * obj + log1pf(expf(-fabsf(l0)));
  }

  // ---- phase 3: WMMA wave reduction -> LDS float atomics -> global atomics ----
  const float rb = wave_partial_sum(bce_t);
  const float rx = wave_partial_sum(box_t);
  const float rn = wave_partial_sum(npos_t);
  if (RED_GUARD(tid)) {
    atomicAdd(&s_acc[0], rb);
    atomicAdd(&s_acc[1], rx);
    atomicAdd(&s_acc[2], rn);
  }
  __syncthreads();
  if (tid == 0) {
    const float np = s_acc[2];
    atomicAdd(&acc[0], s_acc[0]);                       // total BCE
    atomicAdd(&acc[1], s_acc[1] / fmaxf(np, 1.0f));     // per-(image,scale) box loss
    atomicAdd(&acc[2], np);                             // total positives
    atomicAdd(&acc[3], np > 0.0f ? 1.0f : 0.0f);        // n_items
  }
}

__global__ void det_loss_final(const float* __restrict__ acc,
                               float* __restrict__ out) {
  const float total_obj = acc[0] / fmaxf(acc[2], 1.0f);
  const float total_box = acc[1] / fmaxf(acc[3], 1.0f);
  out[0] = 1.0f * total_obj + 5.0f * total_box;
}

extern "C" void kernel_launch(void* const* d_in, const int* in_sizes, int n_in,
                              void* d_out, int out_size, void* d_ws, size_t ws_size,
                              hipStream_t stream) {
  const float* p0 = (const float*)d_in[0];
  const float* p1 = (const float*)d_in[1];
  const float* p2 = (const float*)d_in[2];
  const float* gt = (const float*)d_in[3];
  float* acc = (float*)d_ws;           // 4 floats of scratch accumulators
  const int B = in_sizes[3] / (NGT * 4);

  det_loss_init<<<1, 32, 0, stream>>>(acc);
  dim3 grid((unsigned)B, 3);
  det_loss_main<<<grid, BLOCK, 0, stream>>>(p0, p1, p2, gt, acc);
  det_loss_final<<<1, 1, 0, stream>>>(acc, (float*)d_out);
}